// LightGCN_38860864094665
// MI455X (gfx1250) — compile-verified
//
#include <hip/hip_runtime.h>
#include <hip/hip_bf16.h>
#include <stdint.h>

#define NUM_USERS  100001
#define NUM_ITEMS  50001
#define N_NODES    150002
#define EMB        64
#define NNZ_TOT    3200000
#define NUM_LAYERS 3
#define BATCH      4096
#define KITEMS     100

typedef __attribute__((ext_vector_type(2))) float v2f;
typedef __attribute__((ext_vector_type(8))) float v8f;

// x = acc = concat(user_emb, item_emb); row = 64 floats = 16 float4 (aligned)
__global__ void init_concat(const float4* __restrict__ ue, const float4* __restrict__ ie,
                            float4* __restrict__ x, float4* __restrict__ acc, size_t n4) {
    size_t i = (size_t)blockIdx.x * blockDim.x + threadIdx.x;
    if (i >= n4) return;
    const size_t nU4 = (size_t)NUM_USERS * (EMB / 4);
    float4 v = (i < nU4) ? ue[i] : ie[i - nU4];
    x[i] = v;
    acc[i] = v;
}

__global__ void zero_buf(float4* __restrict__ p, size_t n4) {
    size_t i = (size_t)blockIdx.x * blockDim.x + threadIdx.x;
    if (i < n4) p[i] = make_float4(0.f, 0.f, 0.f, 0.f);
}

__global__ void acc_add(float4* __restrict__ acc, const float4* __restrict__ x, size_t n4) {
    size_t i = (size_t)blockIdx.x * blockDim.x + threadIdx.x;
    if (i >= n4) return;
    float4 a = acc[i], b = x[i];
    a.x += b.x; a.y += b.y; a.z += b.z; a.w += b.w;
    acc[i] = a;
}

// One wave32 per nnz: lane handles 2 of 64 dims (float2). Gathers + f32 atomic
// scatter both land in L2 (38.4 MB matrix << 192 MB L2).
__global__ void spmm_scatter(const float* __restrict__ x, float* __restrict__ y,
                             const float* __restrict__ vals, const int* __restrict__ rows,
                             const int* __restrict__ cols, int nnz) {
    int gid  = blockIdx.x * blockDim.x + threadIdx.x;
    int e    = gid >> 5;
    int lane = threadIdx.x & 31;
    if (e >= nnz) return;
    int   c = cols[e];
    int   r = rows[e];
    float v = vals[e];
    v2f src = *(const v2f*)(x + (size_t)c * EMB + lane * 2);
    float* dst = y + (size_t)r * EMB + lane * 2;
    unsafeAtomicAdd(dst,     src.x * v);
    unsafeAtomicAdd(dst + 1, src.y * v);
}

// logits[b,k] = dot(acc_user[users[b]], acc_item[items[b,k]]) / 16
// One wave per (b, 16-item chunk). V_WMMA_F32_16X16X4_F32:
//   A (16x4): lanes 0-15 -> M=lane, K={0,1}; lanes 16-31 -> M=lane-16, K={2,3}
//   B (4x16): lanes 0-15 -> K={0,1};         lanes 16-31 -> K={2,3}   (N = lane%16)
// B = user vector replicated across N -> D[m,n] identical over n; read n=0.
__global__ void logits_wmma(const float* __restrict__ acc, const int* __restrict__ users,
                            const int* __restrict__ items, float* __restrict__ out) {
    int b     = blockIdx.x;
    int chunk = threadIdx.x >> 5;      // 0..6 (7 chunks cover 100 items, padded)
    int lane  = threadIdx.x & 31;
    int m     = lane & 15;
    int kb    = (lane >> 4) << 1;      // 0 or 2: this lane's K-pair base

    int u = users[b];
    int iidx = chunk * 16 + m;
    int iidx_c = iidx < KITEMS ? iidx : (KITEMS - 1);   // clamp padded rows in-bounds
    int it = items[b * KITEMS + iidx_c];

    const float* arow = acc + (size_t)(NUM_USERS + it) * EMB;  // item embedding row
    const float* brow = acc + (size_t)u * EMB;                 // user embedding row

    v8f c = {0.f, 0.f, 0.f, 0.f, 0.f, 0.f, 0.f, 0.f};
#pragma unroll
    for (int k = 0; k < 16; ++k) {                 // 16 chunks of K=4 cover EMB=64
        int off = k * 4 + kb;                      // 8-byte aligned within 256B row
        v2f a  = *(const v2f*)(arow + off);
        v2f bb = *(const v2f*)(brow + off);
        c = __builtin_amdgcn_wmma_f32_16x16x4_f32(
                /*neg_a=*/false, a, /*neg_b=*/false, bb,
                /*c_mod=*/(short)0, c, /*reuse_a=*/false, /*reuse_b=*/false);
    }

    // D layout: lane<16 -> n=lane, VGPR r holds M=r; lane>=16 -> n=lane-16, M=8+r.
    // Take column n==0: lane 0 (M 0..7) and lane 16 (M 8..15).
    if ((lane & 15) == 0) {
        int mbase = chunk * 16 + ((lane >> 4) ? 8 : 0);
#pragma unroll
        for (int r = 0; r < 8; ++r) {
            int idx = mbase + r;
            if (idx < KITEMS)
                out[(size_t)b * KITEMS + idx] = c[r] * 0.0625f;  // (/4)*(/4)
        }
    }
}

extern "C" void kernel_launch(void* const* d_in, const int* in_sizes, int n_in,
                              void* d_out, int out_size, void* d_ws, size_t ws_size,
                              hipStream_t stream) {
    const float* user_emb = (const float*)d_in[0];
    const float* item_emb = (const float*)d_in[1];
    const float* vals     = (const float*)d_in[2];
    const int*   users    = (const int*)d_in[3];
    const int*   items    = (const int*)d_in[4];
    const int*   rows     = (const int*)d_in[5];
    const int*   cols     = (const int*)d_in[6];
    float*       out      = (float*)d_out;

    const size_t n  = (size_t)N_NODES * EMB;   // 9,600,128 floats / buffer
    const size_t n4 = n / 4;
    float* x0  = (float*)d_ws;
    float* x1  = x0 + n;
    float* acc = x1 + n;

    dim3 blk(256);
    dim3 grid_elem((unsigned)((n4 + 255) / 256));

    init_concat<<<grid_elem, blk, 0, stream>>>((const float4*)user_emb,
                                               (const float4*)item_emb,
                                               (float4*)x0, (float4*)acc, n4);

    float* xp = x0;
    float* xn = x1;
    int spmm_blocks = (NNZ_TOT * 32 + 255) / 256;   // one wave per nnz
    for (int l = 0; l < NUM_LAYERS; ++l) {
        zero_buf<<<grid_elem, blk, 0, stream>>>((float4*)xn, n4);
        spmm_scatter<<<spmm_blocks, blk, 0, stream>>>(xp, xn, vals, rows, cols, NNZ_TOT);
        acc_add<<<grid_elem, blk, 0, stream>>>((float4*)acc, (const float4*)xn, n4);
        float* t = xp; xp = xn; xn = t;
    }

    // 7 waves per block (one per 16-item chunk), one block per batch element.
    logits_wmma<<<BATCH, dim3(7 * 32), 0, stream>>>(acc, users, items, out);
}